// RuleAwareAttentionFlow_56934086476224
// MI455X (gfx1250) — compile-verified
//
#include <hip/hip_runtime.h>
#include <hip/hip_bf16.h>

// ---------------------------------------------------------------------------
// RuleAwareAttentionFlow for MI455X (gfx1250, wave32, WMMA)
//
// score(left,right) = leftT * (Wq^T Wk) * right  decomposed into 64-dim blocks
// so the heavy math becomes dense f32 WMMA GEMMs; edge passes are gathered
// dot products + atomics (bandwidth bound on 23.3 TB/s HBM).
// ---------------------------------------------------------------------------

typedef float v2f __attribute__((ext_vector_type(2)));
typedef float v8f __attribute__((ext_vector_type(8)));

__device__ __forceinline__ unsigned f2ord(float f) {
    unsigned u = __float_as_uint(f);
    return (u & 0x80000000u) ? ~u : (u | 0x80000000u);
}
__device__ __forceinline__ float ord2f(unsigned u) {
    return (u & 0x80000000u) ? __uint_as_float(u & 0x7FFFFFFFu)
                             : __uint_as_float(~u);
}

// ---------------------------------------------------------------------------
// Generic f32 WMMA GEMM: C[M,N] = A[M,K] @ B[K,N]   (row-major, all dims %16/%4)
// FLAGS bit0: A is transposed in memory (A is K x M, used for Wq^T @ Wk)
// FLAGS bit1: epilogue  C = leaky_relu(C + bias[n], 0.01)
// One wave computes one 16x16 C tile via v_wmma_f32_16x16x4_f32.
// ---------------------------------------------------------------------------
template <int FLAGS>
__global__ __launch_bounds__(256) void gemm_wmma_f32(
    const float* __restrict__ A, int lda,
    const float* __restrict__ B, int ldb,
    float* __restrict__ C, int ldc,
    int M, int N, int K,
    const float* __restrict__ bias)
{
    const int wave = blockIdx.x * (blockDim.x >> 5) + (threadIdx.x >> 5);
    const int lane = threadIdx.x & 31;
    const int tilesN = N >> 4;
    const int tm = wave / tilesN;
    const int tn = wave - tm * tilesN;
    if (tm * 16 >= M) return;               // uniform per wave -> EXEC all-1s
    const int m0 = tm << 4, n0 = tn << 4;
    const int row = lane & 15;              // A row (or C col) within tile
    const int hi  = lane >> 4;              // K-half selector

    v8f c = {0.f, 0.f, 0.f, 0.f, 0.f, 0.f, 0.f, 0.f};
    for (int k0 = 0; k0 < K; k0 += 4) {
        const int ka = k0 + 2 * hi;
        v2f a, b;
        if (FLAGS & 1) {                    // A stored K x M
            a.x = A[(size_t)ka * lda + (m0 + row)];
            a.y = A[(size_t)(ka + 1) * lda + (m0 + row)];
        } else {                            // A stored M x K
            const float* ap = A + (size_t)(m0 + row) * lda + ka;
            a.x = ap[0];
            a.y = ap[1];
        }
        b.x = B[(size_t)ka * ldb + (n0 + row)];
        b.y = B[(size_t)(ka + 1) * ldb + (n0 + row)];
        c = __builtin_amdgcn_wmma_f32_16x16x4_f32(
                false, a, false, b, (short)0, c, false, false);
    }
#pragma unroll
    for (int r = 0; r < 8; ++r) {
        const int m = m0 + r + 8 * hi;
        const int n = n0 + row;
        float v = c[r];
        if (FLAGS & 2) {
            v += bias[n];
            v = v > 0.f ? v : 0.01f * v;
        }
        C[(size_t)m * ldc + n] = v;
    }
}

// ---------------------------------------------------------------------------
// Assemble W3 (64x192) = [M00 | M01 | M10^T] and WlinT (64x64) = W_lin^T
// ---------------------------------------------------------------------------
__global__ void prep_weights(const float* __restrict__ M256,
                             const float* __restrict__ Wlin,
                             float* __restrict__ W3,
                             float* __restrict__ WlinT)
{
    const int g = blockIdx.x * blockDim.x + threadIdx.x;
    if (g < 64 * 192) {
        const int a = g / 192, cIdx = g - a * 192;
        W3[g] = (cIdx < 128) ? M256[a * 256 + cIdx]
                             : M256[(cIdx - 64) * 256 + a];   // M10^T
    }
    if (g < 64 * 64) {
        const int a = g >> 6, b = g & 63;
        WlinT[g] = Wlin[b * 64 + a];
    }
}

// ---------------------------------------------------------------------------
// Per-query vectors: qvec[q] = [wq(64) | tq(64) | uq(64)], cq[q] scalar.
// One 64-thread block per query; M256 blocks streamed from L2 (256KB, hot).
// ---------------------------------------------------------------------------
__global__ __launch_bounds__(64) void query_prep(
    const float* __restrict__ qst_tab, const float* __restrict__ qr_tab,
    const float* __restrict__ M, float* __restrict__ qvec,
    float* __restrict__ cqv)
{
    const int q = blockIdx.x, t = threadIdx.x;
    __shared__ float x[128];
    __shared__ float red[64];
    x[t]      = qst_tab[q * 64 + t];
    x[64 + t] = qr_tab[q * 64 + t];
    __syncthreads();

    float w = 0.f, tt = 0.f, u = 0.f;
    for (int g = 0; g < 64; ++g) {
        const float s = x[g], r = x[64 + g];
        w  += s * (M[(64 + t) * 256 + 128 + g] + M[(128 + g) * 256 + 64 + t]);
        w  += r * (M[(64 + t) * 256 + 192 + g] + M[(192 + g) * 256 + 64 + t]);
        tt += s * M[t * 256 + 128 + g] + r * M[t * 256 + 192 + g];
        u  += s * M[(128 + g) * 256 + t] + r * M[(192 + g) * 256 + t];
    }
    qvec[(size_t)q * 192 + t]       = w;
    qvec[(size_t)q * 192 + 64 + t]  = tt;
    qvec[(size_t)q * 192 + 128 + t] = u;

    float p = 0.f;
    const float xt = x[t], xt64 = x[64 + t];
    for (int d = 0; d < 128; ++d)
        p += x[d] * (xt * M[(128 + t) * 256 + 128 + d] +
                     xt64 * M[(192 + t) * 256 + 128 + d]);
    red[t] = p;
    __syncthreads();
    if (t == 0) {
        float s = 0.f;
        for (int i = 0; i < 64; ++i) s += red[i];
        cqv[q] = s;
    }
}

// ---------------------------------------------------------------------------
// Per-edge logits: one wave per edge, lane handles 2 dims (float2), shuffle
// reduce, then ordered-uint atomicMax for the segment max.
// logit = (P0[src]+uq).H[dst] + tq.H[src] + (P1[src]+PD[dst]+R1[e]+wq).rel
//         + cq + gamma*prior
// ---------------------------------------------------------------------------
__global__ __launch_bounds__(256) void edge_logits(
    const int* __restrict__ edges, int nE,
    const float* __restrict__ rel, const float* __restrict__ R1,
    const float* __restrict__ P, const float* __restrict__ H,
    const float* __restrict__ qvec, const float* __restrict__ cqv,
    const float* __restrict__ prior, const float* __restrict__ gamma_p,
    float* __restrict__ logits, unsigned* __restrict__ smax)
{
    const int w = (blockIdx.x * blockDim.x + threadIdx.x) >> 5;
    const int lane = threadIdx.x & 31;
    if (w >= nE) return;
    const int* ed = edges + (size_t)w * 8;
    const int qid = ed[0], src = ed[6], dst = ed[7];

    const float2 R  = ((const float2*)(rel + (size_t)w * 64))[lane];
    const float2 R1v= ((const float2*)(R1  + (size_t)w * 64))[lane];
    const float2 P0 = ((const float2*)(P + (size_t)src * 192))[lane];
    const float2 P1 = ((const float2*)(P + (size_t)src * 192 + 64))[lane];
    const float2 PD = ((const float2*)(P + (size_t)dst * 192 + 128))[lane];
    const float2 HS = ((const float2*)(H + (size_t)src * 64))[lane];
    const float2 HD = ((const float2*)(H + (size_t)dst * 64))[lane];
    const float2 WQ = ((const float2*)(qvec + (size_t)qid * 192))[lane];
    const float2 TQ = ((const float2*)(qvec + (size_t)qid * 192 + 64))[lane];
    const float2 UQ = ((const float2*)(qvec + (size_t)qid * 192 + 128))[lane];

    float acc = (P0.x + UQ.x) * HD.x + (P0.y + UQ.y) * HD.y
              + TQ.x * HS.x + TQ.y * HS.y
              + (P1.x + PD.x + R1v.x + WQ.x) * R.x
              + (P1.y + PD.y + R1v.y + WQ.y) * R.y;
    for (int off = 16; off; off >>= 1) acc += __shfl_xor(acc, off, 32);

    if (lane == 0) {
        float lg = acc + cqv[qid];
        if (prior) lg += gamma_p[0] * prior[w];
        logits[w] = lg;
        atomicMax(smax + src, f2ord(lg));
    }
}

// ---------------------------------------------------------------------------
// Segment softmax pass 2/3 (thread per edge).
// ---------------------------------------------------------------------------
__global__ void edge_exp(const int* __restrict__ edges, int nE,
                         const float* __restrict__ logits,
                         const unsigned* __restrict__ smax,
                         float* __restrict__ trans,
                         float* __restrict__ ssum)
{
    const int e = blockIdx.x * blockDim.x + threadIdx.x;
    if (e >= nE) return;
    const int src = edges[(size_t)e * 8 + 6];
    const float v = __expf(logits[e] - ord2f(smax[src]));
    trans[e] = v;
    atomicAdd(ssum + src, v);
}

__global__ void edge_norm(const int* __restrict__ edges, int nE,
                          const float* __restrict__ ssum,
                          const float* __restrict__ visited,
                          float* __restrict__ trans,
                          float* __restrict__ ts)
{
    const int e = blockIdx.x * blockDim.x + threadIdx.x;
    if (e >= nE) return;
    const int src = edges[(size_t)e * 8 + 6];
    const float t = trans[e] / ssum[src];
    trans[e] = t;
    if (ts) ts[e] = t * visited[src];
}

// ---------------------------------------------------------------------------
// Top-K per query over its 64 contiguous edges (wave per query).
// Iterated butterfly argmax; jax.lax.top_k tie-break: larger value, then
// lower index, emitted in descending order. Also scatters the pruned
// node-score contribution.
// ---------------------------------------------------------------------------
__global__ __launch_bounds__(32) void topk_prune(
    const float* __restrict__ ts, const float* __restrict__ trans,
    const int* __restrict__ edges1,
    int* __restrict__ out_pruned, int* __restrict__ out_orig,
    float* __restrict__ transp, float* __restrict__ out_score, int K)
{
    const int q = blockIdx.x, lane = threadIdx.x;
    const float* rowp = ts + (size_t)q * 64;
    const float v0 = rowp[lane], v1 = rowp[lane + 32];
    bool s0 = false, s1 = false;

    for (int it = 0; it < K; ++it) {
        float a0 = s0 ? -1e30f : v0;
        float a1 = s1 ? -1e30f : v1;
        float cv; int ci;
        if (a0 >= a1) { cv = a0; ci = lane; } else { cv = a1; ci = lane + 32; }
        for (int off = 16; off; off >>= 1) {
            const float ov = __shfl_xor(cv, off, 32);
            const int   oi = __shfl_xor(ci, off, 32);
            if (ov > cv || (ov == cv && oi < ci)) { cv = ov; ci = oi; }
        }
        if (lane == (ci & 31)) { if (ci < 32) s0 = true; else s1 = true; }
        if (lane == 0) {
            const int oi = q * 64 + ci;
            const int o  = q * K + it;
            out_orig[o] = oi;
            transp[o]   = trans[oi];
            for (int c = 0; c < 8; ++c)
                out_pruned[(size_t)o * 8 + c] = edges1[(size_t)oi * 8 + c];
            atomicAdd(out_score + edges1[(size_t)oi * 8 + 7], cv);
        }
    }
}

// ---------------------------------------------------------------------------
// agg[src] += t * H[dst]  (wave per edge, float2 lanes), cnt[src]++.
// ---------------------------------------------------------------------------
__global__ __launch_bounds__(256) void agg_edges(
    const int* __restrict__ edges, int nE,
    const float* __restrict__ trans, const float* __restrict__ H,
    float* __restrict__ agg, unsigned* __restrict__ cnt)
{
    const int w = (blockIdx.x * blockDim.x + threadIdx.x) >> 5;
    const int lane = threadIdx.x & 31;
    if (w >= nE) return;
    const int src = edges[(size_t)w * 8 + 6];
    const int dst = edges[(size_t)w * 8 + 7];
    const float t = trans[w];
    const float2 hv = ((const float2*)(H + (size_t)dst * 64))[lane];
    atomicAdd(agg + (size_t)src * 64 + 2 * lane,     t * hv.x);
    atomicAdd(agg + (size_t)src * 64 + 2 * lane + 1, t * hv.y);
    if (lane == 0) atomicAdd(cnt + src, 1u);
}

// out[i,:] = cnt[i] ? agg[i,:] : fallback[i,:]   (RATIO == 0)
__global__ void apply_update(const float* __restrict__ agg,
                             const unsigned* __restrict__ cnt,
                             const float* __restrict__ fallback,
                             float* __restrict__ out, int total)
{
    const int g = blockIdx.x * blockDim.x + threadIdx.x;
    if (g >= total) return;
    out[g] = cnt[g >> 6] ? agg[g] : fallback[g];
}

// ---------------------------------------------------------------------------
extern "C" void kernel_launch(void* const* d_in, const int* in_sizes, int n_in,
                              void* d_out, int out_size, void* d_ws, size_t ws_size,
                              hipStream_t stream)
{
    const float* visited   = (const float*)d_in[0];
    const float* node_repr = (const float*)d_in[1];
    const float* rel0      = (const float*)d_in[2];
    const float* rel1      = (const float*)d_in[3];
    const float* prior     = (const float*)d_in[4];
    const float* qst       = (const float*)d_in[5];
    const float* qre       = (const float*)d_in[6];
    const float* Wq        = (const float*)d_in[7];
    const float* Wk        = (const float*)d_in[8];
    const float* Wlin      = (const float*)d_in[9];
    const float* blin      = (const float*)d_in[10];
    const float* gamma     = (const float*)d_in[11];
    const int*   edges0    = (const int*)d_in[12];
    const int*   edges1    = (const int*)d_in[13];

    const int Nn = in_sizes[0];
    const int E  = in_sizes[4];
    const int E0 = in_sizes[2] / 64;
    const int Q  = in_sizes[5] / 64;
    const long long rem = (long long)out_size - (long long)Nn * 65;
    const int K  = (int)(rem / (9LL * Q));     // = 10 with the reference sizes
    const int QK = Q * K;

    float* out_score = (float*)d_out;
    float* out_repr  = out_score + Nn;
    int*   out_pruned = (int*)(out_repr + (size_t)Nn * 64);
    int*   out_orig   = out_pruned + (size_t)QK * 8;

    // ---- workspace carve ----
    float* p = (float*)d_ws;
    auto take = [&](size_t n) { float* r = p; p += n; return r; };
    const int Emax = E > E0 ? E : E0;
    float*    M256   = take(256 * 256);
    float*    W3     = take(64 * 192);
    float*    WlinT  = take(64 * 64);
    float*    qvec   = take((size_t)Q * 192);
    float*    cqv    = take(Q);
    float*    P      = take((size_t)Nn * 192);
    float*    R1a    = take((size_t)E * 64);
    float*    R1b    = take((size_t)E0 * 64);
    float*    logits = take((size_t)Emax);
    float*    trans  = take((size_t)Emax);
    float*    ts     = take((size_t)E);
    unsigned* smax   = (unsigned*)take(Nn);
    float*    ssum   = take(Nn);
    unsigned* cnt    = (unsigned*)take(Nn);
    float*    agg    = take((size_t)Nn * 64);
    float*    upd1   = take((size_t)Nn * 64);
    float*    transp = take((size_t)QK);
    if ((size_t)((char*)p - (char*)d_ws) > ws_size) return;   // ws too small

    const int TPB = 256;
    auto gemmBlocks = [](int M, int N) { return ((M >> 4) * (N >> 4) + 7) / 8; };
    auto waveBlocks = [](int n) { return (n + 7) / 8; };
    auto thrBlocks  = [](int n) { return (n + 255) / 256; };

    // ---- init ----
    hipMemsetAsync(smax, 0, (size_t)Nn * 4, stream);
    hipMemsetAsync(ssum, 0, (size_t)Nn * 4, stream);
    hipMemsetAsync(cnt,  0, (size_t)Nn * 4, stream);
    hipMemsetAsync(agg,  0, (size_t)Nn * 256, stream);
    hipMemsetAsync(out_score, 0, (size_t)Nn * 4, stream);

    // ---- dense precomputes (WMMA) ----
    // M256 = Wq^T @ Wk
    gemm_wmma_f32<1><<<gemmBlocks(256, 256), TPB, 0, stream>>>(
        Wq, 256, Wk, 256, M256, 256, 256, 256, 256, nullptr);
    prep_weights<<<(64 * 192 + 255) / 256, TPB, 0, stream>>>(M256, Wlin, W3, WlinT);
    query_prep<<<Q, 64, 0, stream>>>(qst, qre, M256, qvec, cqv);
    // P = node_repr @ [M00|M01|M10^T]
    gemm_wmma_f32<0><<<gemmBlocks(Nn, 192), TPB, 0, stream>>>(
        node_repr, 64, W3, 192, P, 192, Nn, 192, 64, nullptr);
    // R1 = rel_emb @ M11 (block at row 64, col 64 of M256)
    gemm_wmma_f32<0><<<gemmBlocks(E, 64), TPB, 0, stream>>>(
        rel1, 64, M256 + 64 * 256 + 64, 256, R1a, 64, E, 64, 64, nullptr);
    gemm_wmma_f32<0><<<gemmBlocks(E0, 64), TPB, 0, stream>>>(
        rel0, 64, M256 + 64 * 256 + 64, 256, R1b, 64, E0, 64, 64, nullptr);

    // ---- stage 1: edges1 attention, softmax over src segments ----
    edge_logits<<<waveBlocks(E), TPB, 0, stream>>>(
        edges1, E, rel1, R1a, P, node_repr, qvec, cqv, prior, gamma,
        logits, smax);
    edge_exp<<<thrBlocks(E), TPB, 0, stream>>>(edges1, E, logits, smax, trans, ssum);
    edge_norm<<<thrBlocks(E), TPB, 0, stream>>>(edges1, E, ssum, visited, trans, ts);

    // ---- top-K pruning + node score scatter ----
    topk_prune<<<Q, 32, 0, stream>>>(ts, trans, edges1,
                                     out_pruned, out_orig, transp, out_score, K);

    // ---- stage 1 repr update ----
    agg_edges<<<waveBlocks(QK), TPB, 0, stream>>>(out_pruned, QK, transp,
                                                  node_repr, agg, cnt);
    apply_update<<<thrBlocks(Nn * 64), TPB, 0, stream>>>(agg, cnt, node_repr,
                                                         upd1, Nn * 64);

    // ---- stage 2: edges0 attention on updated repr ----
    hipMemsetAsync(smax, 0, (size_t)Nn * 4, stream);
    hipMemsetAsync(ssum, 0, (size_t)Nn * 4, stream);
    hipMemsetAsync(cnt,  0, (size_t)Nn * 4, stream);
    hipMemsetAsync(agg,  0, (size_t)Nn * 256, stream);

    gemm_wmma_f32<0><<<gemmBlocks(Nn, 192), TPB, 0, stream>>>(
        upd1, 64, W3, 192, P, 192, Nn, 192, 64, nullptr);
    edge_logits<<<waveBlocks(E0), TPB, 0, stream>>>(
        edges0, E0, rel0, R1b, P, upd1, qvec, cqv, nullptr, gamma,
        logits, smax);
    edge_exp<<<thrBlocks(E0), TPB, 0, stream>>>(edges0, E0, logits, smax, trans, ssum);
    edge_norm<<<thrBlocks(E0), TPB, 0, stream>>>(edges0, E0, ssum, nullptr, trans,
                                                 nullptr);
    agg_edges<<<waveBlocks(E0), TPB, 0, stream>>>(edges0, E0, trans, upd1, agg, cnt);
    apply_update<<<thrBlocks(Nn * 64), TPB, 0, stream>>>(agg, cnt, upd1,
                                                         agg, Nn * 64); // in place

    // ---- final linear + leaky relu (WMMA, fused epilogue) ----
    gemm_wmma_f32<2><<<gemmBlocks(Nn, 64), TPB, 0, stream>>>(
        agg, 64, WlinT, 64, out_repr, 64, Nn, 64, 64, blin);
}